// SimpleGatedUnit2_13795434954910
// MI455X (gfx1250) — compile-verified
//
#include <hip/hip_runtime.h>
#include <hip/hip_bf16.h>
#include <math.h>

typedef float v2f __attribute__((ext_vector_type(2)));
typedef float v8f __attribute__((ext_vector_type(8)));

// D = A(16x4 f32) * B(4x16 f32) + C(16x16 f32), wave32, fp32 WMMA path
__device__ __forceinline__ v8f wmma_f32_4(v2f a, v2f b, v8f c) {
    return __builtin_amdgcn_wmma_f32_16x16x4_f32(
        /*neg_a=*/false, a, /*neg_b=*/false, b,
        /*c_mod=*/(short)0, c, /*reuse_a=*/false, /*reuse_b=*/false);
}

__device__ __forceinline__ float hard_sigmoid_f(float x) {
    return fminf(fmaxf(0.2f * x + 0.5f, 0.0f), 1.0f);
}
__device__ __forceinline__ float softplus_f(float x) {
    return fmaxf(x, 0.0f) + log1pf(expf(-fabsf(x)));   // stable softplus
}

// ---------------------------------------------------------------------------
// Kernel 0: transpose [R,128] -> [128,R] so WMMA B-fragments (two K-adjacent
// values at fixed N) become one contiguous global_load_b64.
// ---------------------------------------------------------------------------
__global__ __launch_bounds__(256) void sgu_transpose128_kernel(
    const float* __restrict__ src, float* __restrict__ dst, int R)
{
    const int i = blockIdx.x * 256 + threadIdx.x;   // grid sized exactly R*128/256
    const int r = i >> 7;
    const int c = i & 127;
    dst[c * R + r] = src[i];
}

// ---------------------------------------------------------------------------
// Kernel 1: projection  xx = X@W + b, xg = X@W_gate + b_gate
// Block = 256 threads (8 waves) owns a 32-row M-block (2 WMMA M-tiles) of the
// flattened [B*T, D] X; wave w owns N-tile w of H=128.  X block staged in LDS;
// B-fragments are b64 loads from the transposed (L2-hot) weights, amortized
// over both M-tiles.  Output layout [T,B,H] for contiguous per-step scan reads.
// ---------------------------------------------------------------------------
__global__ __launch_bounds__(256) void sgu_proj_kernel(
    const float* __restrict__ X,
    const float* __restrict__ Wt,  const float* __restrict__ Wgt,  // [H,D]
    const float* __restrict__ bb,  const float* __restrict__ bg,
    float* __restrict__ xx, float* __restrict__ xg, int tshift)
{
    constexpr int D = 256, H = 128, B = 64, ROWS = 32;
    constexpr int LDX = 260;                 // 260 % 64 == 4 -> bank-skewed rows
    __shared__ float Xs[ROWS * LDX];         // 33,280 B

    const int tid  = threadIdx.x;
    const int wave = tid >> 5;
    const int lane = tid & 31;
    const int half = lane >> 4;
    const int r    = lane & 15;
    const int m0   = blockIdx.x * ROWS;      // global row block in [B*T]
    const int n0   = wave * 16;              // this wave's N-tile

    // Cooperative X stage: 32 rows x 64 float4 = 2048 vec4 (8 per thread)
    for (int i = tid; i < ROWS * (D / 4); i += 256) {
        const int row = i >> 6;
        const int c4  = i & 63;
        const float4 v = ((const float4*)(X + (size_t)(m0 + row) * D))[c4];
        float* dst = &Xs[row * LDX + c4 * 4];
        dst[0] = v.x; dst[1] = v.y; dst[2] = v.z; dst[3] = v.w;
    }
    __syncthreads();

    v8f acc00 = {}, acc01 = {};              // M-tile 0: @W, @Wg
    v8f acc10 = {}, acc11 = {};              // M-tile 1: @W, @Wg
    const float* wrow  = Wt  + (size_t)(n0 + r) * D;
    const float* wgrow = Wgt + (size_t)(n0 + r) * D;
    #pragma unroll 4
    for (int k = 0; k < D; k += 4) {
        const int ka = k + 2 * half;         // lanes 0-15: K={k,k+1}; 16-31: K={k+2,k+3}
        const v2f a0 = *(const v2f*)&Xs[r        * LDX + ka];
        const v2f a1 = *(const v2f*)&Xs[(16 + r) * LDX + ka];
        const v2f bw = *(const v2f*)&wrow[ka];   // {W[ka][n], W[ka+1][n]}
        const v2f bv = *(const v2f*)&wgrow[ka];
        acc00 = wmma_f32_4(a0, bw, acc00);
        acc10 = wmma_f32_4(a1, bw, acc10);
        acc01 = wmma_f32_4(a0, bv, acc01);
        acc11 = wmma_f32_4(a1, bv, acc11);
    }

    const float bias0 = bb[n0 + r];
    const float bias1 = bg[n0 + r];
    const int   tmask = (1 << tshift) - 1;
    #pragma unroll
    for (int v = 0; v < 8; ++v) {
        const int M = v + 8 * half;          // C/D layout: VGPR v -> M = v (+8 hi lanes)
        #pragma unroll
        for (int mt = 0; mt < 2; ++mt) {
            const int grow = m0 + mt * 16 + M;
            const int bi   = grow >> tshift;         // batch   (row = b*T + t)
            const int ti   = grow & tmask;           // timestep
            const size_t o = (size_t)ti * B * H + (size_t)bi * H + n0 + r;
            const v8f& aX = mt ? acc10 : acc00;
            const v8f& aG = mt ? acc11 : acc01;
            xx[o] = aX[v] + bias0;
            xg[o] = aG[v] + bias1;
        }
    }
}

// ---------------------------------------------------------------------------
// Kernel 2: sequential scan over T steps.  Recurrence is independent per batch
// row -> 4 workgroups each own a 16-row batch slice (one WMMA M-tile); 8 waves
// cover the 8 N-tiles of H.  h double-buffered in LDS; U/Ug fragments are b64
// loads from the transposed copies (L2/L0-resident).
// ---------------------------------------------------------------------------
__global__ __launch_bounds__(256) void sgu_scan_kernel(
    const float* __restrict__ xx, const float* __restrict__ xg,
    const float* __restrict__ Ut, const float* __restrict__ Ugt,  // [H,H] transposed
    float* __restrict__ out, int T)
{
    constexpr int H = 128, B = 64;
    constexpr int LDH = 132;                 // 132 % 64 == 4 -> bank-skewed rows
    __shared__ float hbuf[2][16 * LDH];
    __shared__ float gbuf[16 * LDH];

    const int tid  = threadIdx.x;
    const int wave = tid >> 5;
    const int lane = tid & 31;
    const int half = lane >> 4;
    const int r    = lane & 15;
    const int b0   = blockIdx.x * 16;
    const int n0   = wave * 16;

    for (int i = tid; i < 2 * 16 * LDH; i += 256) (&hbuf[0][0])[i] = 0.0f;
    __syncthreads();

    const float* urow  = Ut  + (size_t)(n0 + r) * H;
    const float* ugrow = Ugt + (size_t)(n0 + r) * H;

    int cur = 0;
    for (int t = 0; t < T; ++t) {
        const size_t base = (size_t)t * B * H + (size_t)b0 * H;
        const float* __restrict__ xxt = xx + base;
        const float* __restrict__ xgt = xg + base;
        float* hc = hbuf[cur];
        float* hn = hbuf[cur ^ 1];

        if (t + 1 < T) {                     // stream next step's rows toward L0
            __builtin_prefetch(xx + base + (size_t)B * H, 0, 1);
            __builtin_prefetch(xg + base + (size_t)B * H, 0, 1);
        }

        // g = xg_t * h  (coalesced global read, LDS write)
        for (int i = tid; i < 16 * H; i += 256) {
            const int row = i >> 7, col = i & 127;
            gbuf[row * LDH + col] = xgt[row * H + col] * hc[row * LDH + col];
        }
        __syncthreads();

        v8f acc1 = {};  // h @ U
        v8f acc2 = {};  // g @ Ug
        #pragma unroll 4
        for (int k = 0; k < H; k += 4) {
            const int ka = k + 2 * half;
            const v2f a1 = *(const v2f*)&hc[r * LDH + ka];
            const v2f bU = *(const v2f*)&urow[ka];
            acc1 = wmma_f32_4(a1, bU, acc1);
            const v2f a2 = *(const v2f*)&gbuf[r * LDH + ka];
            const v2f bG = *(const v2f*)&ugrow[ka];
            acc2 = wmma_f32_4(a2, bG, acc2);
        }

        // Gating + state update on this wave's disjoint 16x16 tile
        #pragma unroll
        for (int v = 0; v < 8; ++v) {
            const int M = v + 8 * half;
            const float hprev = hc[M * LDH + n0 + r];
            const float z  = hard_sigmoid_f(xxt[M * H + n0 + r] + acc1[v]);
            const float zg = tanhf(acc2[v]);
            const float zo = softplus_f(hprev * zg);
            hn[M * LDH + n0 + r] = z * zo + (1.0f - z) * hprev;
        }
        __syncthreads();
        cur ^= 1;
    }

    const float* hf = hbuf[cur];
    for (int i = tid; i < 16 * H; i += 256) {
        const int row = i >> 7, col = i & 127;
        out[(size_t)(b0 + row) * H + col] = hf[row * LDH + col];
    }
}

extern "C" void kernel_launch(void* const* d_in, const int* in_sizes, int n_in,
                              void* d_out, int out_size, void* d_ws, size_t ws_size,
                              hipStream_t stream) {
    // setup_inputs order: X, W, U, b, W_gate, b_gate, U_gate  (all fp32)
    const float* X  = (const float*)d_in[0];
    const float* W  = (const float*)d_in[1];
    const float* U  = (const float*)d_in[2];
    const float* b  = (const float*)d_in[3];
    const float* Wg = (const float*)d_in[4];
    const float* bg = (const float*)d_in[5];
    const float* Ug = (const float*)d_in[6];

    constexpr int B = 64, D = 256, H = 128;
    const int T = in_sizes[0] / (B * D);     // 2048 (power of two)
    int tshift = 0;
    while ((1 << tshift) < T) ++tshift;

    // Workspace layout
    float* xx  = (float*)d_ws;                         // [T,B,H]
    float* xg  = xx  + (size_t)T * B * H;              // [T,B,H]
    float* Wt  = xg  + (size_t)T * B * H;              // [H,D]
    float* Wgt = Wt  + (size_t)H * D;                  // [H,D]
    float* Ut  = Wgt + (size_t)H * D;                  // [H,H]
    float* Ugt = Ut  + (size_t)H * H;                  // [H,H]
    float* out = (float*)d_out;                        // [B,H]

    // Transpose weights once (tiny; stays hot in 192MB L2)
    sgu_transpose128_kernel<<<dim3((D * H) / 256), dim3(256), 0, stream>>>(W,  Wt,  D);
    sgu_transpose128_kernel<<<dim3((D * H) / 256), dim3(256), 0, stream>>>(Wg, Wgt, D);
    sgu_transpose128_kernel<<<dim3((H * H) / 256), dim3(256), 0, stream>>>(U,  Ut,  H);
    sgu_transpose128_kernel<<<dim3((H * H) / 256), dim3(256), 0, stream>>>(Ug, Ugt, H);

    sgu_proj_kernel<<<dim3((B * T) / 32), dim3(256), 0, stream>>>(
        X, Wt, Wgt, b, bg, xx, xg, tshift);
    sgu_scan_kernel<<<dim3(B / 16), dim3(256), 0, stream>>>(
        xx, xg, Ut, Ugt, out, T);
}